// geo_MLP_53369263620213
// MI455X (gfx1250) — compile-verified
//
#include <hip/hip_runtime.h>

// Problem constants (match reference)
#define NGENE  (128*2048)      // 262144
#define NDRUG  128
#define E_DG_  (NDRUG*256)     // 32768
#define E_GG_  (NGENE*16)      // 4194304
#define FGENE  64
#define FDRUG  128
#define NB     128             // batch B
#define GPG    2048            // genes per graph

// ---------- helpers ----------
static __device__ __forceinline__ unsigned encf(float f) {
  unsigned u = __float_as_uint(f);
  return (u & 0x80000000u) ? ~u : (u | 0x80000000u);   // order-preserving float->uint
}
static __device__ __forceinline__ float decf(unsigned u) {
  return (u & 0x80000000u) ? __uint_as_float(u & 0x7fffffffu) : __uint_as_float(~u);
}
static __device__ __forceinline__ float lrelu02(float x) { return x >= 0.f ? x : 0.2f * x; }
static __device__ __forceinline__ float selu_f(float x) {
  const float kScale = 1.0507009873554805f;
  const float kAlpha = 1.6732632423543772f;
  return x > 0.f ? kScale * x : kScale * kAlpha * (expf(x) - 1.f);
}

// ---------- drug-side projection: hs_dg[d] = dot(x_drug[d], Wdg_src) ----------
__global__ void drug_proj_k(const float* __restrict__ xd, const float* __restrict__ w,
                            float* __restrict__ hs) {
  int d = blockIdx.x * blockDim.x + threadIdx.x;
  if (d >= NDRUG) return;
  const float4* row = (const float4*)(xd + (size_t)d * FDRUG);
  const float4* wc  = (const float4*)w;
  float acc = 0.f;
#pragma unroll
  for (int i = 0; i < FDRUG / 4; ++i) {
    float4 x = row[i], c = wc[i];
    acc += x.x * c.x + x.y * c.y + x.z * c.z + x.w * c.w;
  }
  hs[d] = acc;
}

// ---------- gene-side projections (3 dot products in one x_gene pass) ----------
__global__ void gene_proj_k(const float* __restrict__ xg,
                            const float* __restrict__ wdg_dst, const float* __restrict__ adg_dst,
                            const float* __restrict__ wgg_src,
                            const float* __restrict__ wgg_dst, const float* __restrict__ agg_dst,
                            float* __restrict__ sdst_dg, float* __restrict__ hs_gg,
                            float* __restrict__ sdst_gg) {
  int g = blockIdx.x * blockDim.x + threadIdx.x;
  if (g >= NGENE) return;
  const float4* row = (const float4*)(xg + (size_t)g * FGENE);
  const float4* w0  = (const float4*)wdg_dst;
  const float4* w1  = (const float4*)wgg_src;
  const float4* w2  = (const float4*)wgg_dst;
  float a0 = 0.f, a1 = 0.f, a2 = 0.f;
#pragma unroll
  for (int i = 0; i < FGENE / 4; ++i) {
    float4 x = row[i];
    float4 c0 = w0[i], c1 = w1[i], c2 = w2[i];
    a0 += x.x * c0.x + x.y * c0.y + x.z * c0.z + x.w * c0.w;
    a1 += x.x * c1.x + x.y * c1.y + x.z * c1.z + x.w * c1.w;
    a2 += x.x * c2.x + x.y * c2.y + x.z * c2.z + x.w * c2.w;
  }
  sdst_dg[g] = a0 * adg_dst[0];
  hs_gg[g]   = a1;
  sdst_gg[g] = a2 * agg_dst[0];
}

// ---------- GAT edge pass 1: segment max of leaky_relu scores ----------
// e < E : edge from list ; e >= E : self loop (src = dst = e - E)
__global__ void gat_max_k(const int* __restrict__ src, const int* __restrict__ dst,
                          int E, int nself,
                          const float* __restrict__ hs_src, const float* __restrict__ a_src,
                          const float* __restrict__ s_dst, unsigned* __restrict__ mbuf) {
  int e = blockIdx.x * blockDim.x + threadIdx.x;
  if (e >= E + nself) return;
  int s, d;
  if (e < E) { s = src[e]; d = dst[e]; } else { s = d = e - E; }
  float alpha = lrelu02(hs_src[s] * a_src[0] + s_dst[d]);
  atomicMax(&mbuf[d], encf(alpha));
}

// ---------- GAT edge pass 2: z += e ; num += hs[src]*e ----------
__global__ void gat_acc_k(const int* __restrict__ src, const int* __restrict__ dst,
                          int E, int nself,
                          const float* __restrict__ hs_src, const float* __restrict__ a_src,
                          const float* __restrict__ s_dst, const unsigned* __restrict__ mbuf,
                          float* __restrict__ z, float* __restrict__ num) {
  int e = blockIdx.x * blockDim.x + threadIdx.x;
  if (e >= E + nself) return;
  int s, d;
  if (e < E) { s = src[e]; d = dst[e]; } else { s = d = e - E; }
  float hs = hs_src[s];
  float alpha = lrelu02(hs * a_src[0] + s_dst[d]);
  float ex = __expf(alpha - decf(mbuf[d]));
  atomicAdd(&z[d], ex);
  atomicAdd(&num[d], hs * ex);
}

// ---------- layer-0 combine: h = relu(g_dg + g_gg) ----------
__global__ void combine0_k(const float* __restrict__ num_dg, const float* __restrict__ z_dg,
                           const float* __restrict__ bdg,
                           const float* __restrict__ num_gg, const float* __restrict__ z_gg,
                           const float* __restrict__ bgg, float* __restrict__ h) {
  int g = blockIdx.x * blockDim.x + threadIdx.x;
  if (g >= NGENE) return;
  float a = num_dg[g] / (z_dg[g] + 1e-16f) + bdg[0];
  float b = num_gg[g] / (z_gg[g] + 1e-16f) + bgg[0];
  float v = a + b;
  h[g] = v > 0.f ? v : 0.f;
}

// ---------- layer-1 prep: hs1 = h*W_src ; sdst1 = h*W_dst*a_dst ----------
__global__ void prep1_k(const float* __restrict__ h, const float* __restrict__ w_src,
                        const float* __restrict__ w_dst, const float* __restrict__ a_dst,
                        float* __restrict__ hs1, float* __restrict__ sdst1) {
  int g = blockIdx.x * blockDim.x + threadIdx.x;
  if (g >= NGENE) return;
  float v = h[g];
  hs1[g]   = v * w_src[0];
  sdst1[g] = v * w_dst[0] * a_dst[0];
}

// ---------- layer-1 finish: h2 = relu(num/(z+eps) + b) ----------
__global__ void finish1_k(const float* __restrict__ num, const float* __restrict__ z,
                          const float* __restrict__ b, float* __restrict__ h2) {
  int g = blockIdx.x * blockDim.x + threadIdx.x;
  if (g >= NGENE) return;
  float v = num[g] / (z[g] + 1e-16f) + b[0];
  h2[g] = v > 0.f ? v : 0.f;
}

// ---------- WMMA f32 GEMM: C = act(A[M,K] @ B[K,N] + bias) ----------
// One wave (32 lanes) per 16x16 tile; V_WMMA_F32_16X16X4_F32, K stepped by 4.
// A frag (16x4): lanes 0-15 hold row M=lane, K=0..1 ; lanes 16-31 hold K=2..3.
// B frag (4x16): lanes 0-15 hold col N=lane for K=0..1 ; lanes 16-31 for K=2..3.
// C/D (16x16): lane<16 -> col=lane, rows 0..7 ; lane>=16 -> col=lane-16, rows 8..15.
typedef float v2f __attribute__((ext_vector_type(2)));
typedef float v8f __attribute__((ext_vector_type(8)));

__global__ void gemm_wmma_f32_k(const float* __restrict__ A, const float* __restrict__ B,
                                const float* __restrict__ bias, float* __restrict__ C,
                                int M, int N, int K, int doSelu) {
  int lane = threadIdx.x & 31;
  int half = lane >> 4;
  int l    = lane & 15;
  int n0   = blockIdx.x * 16;
  int m0   = blockIdx.y * 16;

  v8f acc = {};
  const float* Arow = A + (size_t)(m0 + l) * K + 2 * half;   // row m0+l, col base 2*half
  const float* Bcol = B + (size_t)(2 * half) * N + (n0 + l); // row base 2*half, col n0+l

  for (int k = 0; k < K; k += 4) {
    v2f a, b;
    a.x = Arow[k];
    a.y = Arow[k + 1];
    b.x = Bcol[(size_t)k * N];
    b.y = Bcol[(size_t)(k + 1) * N];
    acc = __builtin_amdgcn_wmma_f32_16x16x4_f32(
        /*neg_a=*/false, a, /*neg_b=*/false, b,
        /*c_mod=*/(short)0, acc, /*reuse_a=*/false, /*reuse_b=*/false);
  }

  int col = n0 + l;
  float bv = bias[col];
#pragma unroll
  for (int i = 0; i < 8; ++i) {
    int row = m0 + half * 8 + i;
    float v = acc[i] + bv;
    if (doSelu) v = selu_f(v);
    C[(size_t)row * N + col] = v;
  }
}

// ---------- final linear: out[r] = dot(X[r,0:512], w) + b ----------
__global__ void out_proj_k(const float* __restrict__ X, const float* __restrict__ w,
                           const float* __restrict__ b, float* __restrict__ out) {
  int r = blockIdx.x * blockDim.x + threadIdx.x;
  if (r >= NB) return;
  const float4* row = (const float4*)(X + (size_t)r * 512);
  const float4* wc  = (const float4*)w;
  float acc = 0.f;
#pragma unroll 8
  for (int i = 0; i < 128; ++i) {
    float4 x = row[i], c = wc[i];
    acc += x.x * c.x + x.y * c.y + x.z * c.z + x.w * c.w;
  }
  out[r] = acc + b[0];
}

// ---------- launch ----------
extern "C" void kernel_launch(void* const* d_in, const int* in_sizes, int n_in,
                              void* d_out, int out_size, void* d_ws, size_t ws_size,
                              hipStream_t stream) {
  (void)in_sizes; (void)n_in; (void)out_size; (void)ws_size;

  const float* x_gene   = (const float*)d_in[0];
  const float* x_drug   = (const float*)d_in[1];
  const float* Wdg_src  = (const float*)d_in[2];
  const float* Wdg_dst  = (const float*)d_in[3];
  const float* adg_src  = (const float*)d_in[4];
  const float* adg_dst  = (const float*)d_in[5];
  const float* bdg      = (const float*)d_in[6];
  const float* Wgg0_src = (const float*)d_in[7];
  const float* Wgg0_dst = (const float*)d_in[8];
  const float* agg0_src = (const float*)d_in[9];
  const float* agg0_dst = (const float*)d_in[10];
  const float* bgg0     = (const float*)d_in[11];
  const float* Wgg1_src = (const float*)d_in[12];
  const float* Wgg1_dst = (const float*)d_in[13];
  const float* agg1_src = (const float*)d_in[14];
  const float* agg1_dst = (const float*)d_in[15];
  const float* bgg1     = (const float*)d_in[16];
  const float* mlp_w0   = (const float*)d_in[17];
  const float* mlp_b0   = (const float*)d_in[18];
  const float* mlp_w1   = (const float*)d_in[19];
  const float* mlp_b1   = (const float*)d_in[20];
  const float* mlp_w2   = (const float*)d_in[21];
  const float* mlp_b2   = (const float*)d_in[22];
  const float* mlp_w3   = (const float*)d_in[23];
  const float* mlp_b3   = (const float*)d_in[24];
  const float* mlp_w4   = (const float*)d_in[25];
  const float* mlp_b4   = (const float*)d_in[26];
  const float* out_w    = (const float*)d_in[27];
  const float* out_b    = (const float*)d_in[28];
  const int*   e_dg     = (const int*)d_in[29];   // [2, E_DG] : src row then dst row
  const int*   e_gg     = (const int*)d_in[30];   // [2, E_GG]

  // workspace layout (floats)
  float*    ws       = (float*)d_ws;
  float*    sdst_dg  = ws + 0  * (size_t)NGENE;
  float*    hs_gg0   = ws + 1  * (size_t)NGENE;
  float*    sdst_gg0 = ws + 2  * (size_t)NGENE;
  unsigned* m_a      = (unsigned*)(ws + 3 * (size_t)NGENE); // max buf (dg, then reused layer1)
  float*    z_a      = ws + 4  * (size_t)NGENE;
  float*    num_a    = ws + 5  * (size_t)NGENE;
  unsigned* m_b      = (unsigned*)(ws + 6 * (size_t)NGENE); // max buf (gg0)
  float*    z_b      = ws + 7  * (size_t)NGENE;
  float*    num_b    = ws + 8  * (size_t)NGENE;
  float*    h        = ws + 9  * (size_t)NGENE;
  float*    hs1      = ws + 10 * (size_t)NGENE;
  float*    sdst1    = ws + 11 * (size_t)NGENE;
  float*    h2       = ws + 12 * (size_t)NGENE;             // MLP input, [128, 2048] row-major
  float*    hs_dg    = ws + 13 * (size_t)NGENE;             // 128 floats
  float*    a0 = hs_dg + 256;                               // 128*32
  float*    a1 = a0 + 128 * 32;                             // 128*64
  float*    a2 = a1 + 128 * 64;                             // 128*128
  float*    a3 = a2 + 128 * 128;                            // 128*256
  float*    a4 = a3 + 128 * 256;                            // 128*512

  const int T = 256;
  const int gN   = NGENE / T;              // 1024 blocks over genes
  const int Etot = E_GG_ + NGENE;          // gg edges + self loops

  // zero max/z/num for layer-0 relations (enc ordering makes 0 the floor for maxbuf)
  hipMemsetAsync(ws + 3 * (size_t)NGENE, 0, 6 * (size_t)NGENE * sizeof(float), stream);

  drug_proj_k<<<1, 128, 0, stream>>>(x_drug, Wdg_src, hs_dg);
  gene_proj_k<<<gN, T, 0, stream>>>(x_gene, Wdg_dst, adg_dst, Wgg0_src, Wgg0_dst, agg0_dst,
                                    sdst_dg, hs_gg0, sdst_gg0);

  // drug->gene GAT
  gat_max_k<<<(E_DG_ + T - 1) / T, T, 0, stream>>>(e_dg, e_dg + E_DG_, E_DG_, 0,
                                                   hs_dg, adg_src, sdst_dg, m_a);
  gat_acc_k<<<(E_DG_ + T - 1) / T, T, 0, stream>>>(e_dg, e_dg + E_DG_, E_DG_, 0,
                                                   hs_dg, adg_src, sdst_dg, m_a, z_a, num_a);
  // gene->gene GAT layer 0 (with self loops)
  gat_max_k<<<(Etot + T - 1) / T, T, 0, stream>>>(e_gg, e_gg + E_GG_, E_GG_, NGENE,
                                                  hs_gg0, agg0_src, sdst_gg0, m_b);
  gat_acc_k<<<(Etot + T - 1) / T, T, 0, stream>>>(e_gg, e_gg + E_GG_, E_GG_, NGENE,
                                                  hs_gg0, agg0_src, sdst_gg0, m_b, z_b, num_b);
  combine0_k<<<gN, T, 0, stream>>>(num_a, z_a, bdg, num_b, z_b, bgg0, h);

  // gene->gene GAT layer 1 (features are scalar h)
  hipMemsetAsync(ws + 3 * (size_t)NGENE, 0, 3 * (size_t)NGENE * sizeof(float), stream);
  prep1_k<<<gN, T, 0, stream>>>(h, Wgg1_src, Wgg1_dst, agg1_dst, hs1, sdst1);
  gat_max_k<<<(Etot + T - 1) / T, T, 0, stream>>>(e_gg, e_gg + E_GG_, E_GG_, NGENE,
                                                  hs1, agg1_src, sdst1, m_a);
  gat_acc_k<<<(Etot + T - 1) / T, T, 0, stream>>>(e_gg, e_gg + E_GG_, E_GG_, NGENE,
                                                  hs1, agg1_src, sdst1, m_a, z_a, num_a);
  finish1_k<<<gN, T, 0, stream>>>(num_a, z_a, bgg1, h2);

  // MLP: WMMA f32 GEMMs (one wave per 16x16 tile), SELU epilogue
  dim3 w32(32);
  gemm_wmma_f32_k<<<dim3(32 / 16, 128 / 16),  w32, 0, stream>>>(h2, mlp_w0, mlp_b0, a0, 128, 32, 2048, 1);
  gemm_wmma_f32_k<<<dim3(64 / 16, 128 / 16),  w32, 0, stream>>>(a0, mlp_w1, mlp_b1, a1, 128, 64, 32, 1);
  gemm_wmma_f32_k<<<dim3(128 / 16, 128 / 16), w32, 0, stream>>>(a1, mlp_w2, mlp_b2, a2, 128, 128, 64, 1);
  gemm_wmma_f32_k<<<dim3(256 / 16, 128 / 16), w32, 0, stream>>>(a2, mlp_w3, mlp_b3, a3, 128, 256, 128, 1);
  gemm_wmma_f32_k<<<dim3(512 / 16, 128 / 16), w32, 0, stream>>>(a3, mlp_w4, mlp_b4, a4, 128, 512, 256, 1);

  out_proj_k<<<1, 128, 0, stream>>>(a4, out_w, out_b, (float*)d_out);
}